// GlobalMeanPooling_73461120631369
// MI455X (gfx1250) — compile-verified
//
#include <hip/hip_runtime.h>

// ---------------------------------------------------------------------------
// GlobalMeanPooling on MI455X (gfx1250):
//   out[b, :] = mean(feature[point_idx == b, :])   B=32 segments, D=128 feats
//
// Bandwidth-bound: 1.02 GB single pass => ~44us floor at 23.3 TB/s.
// segment_sum == onehot(point_idx)^T @ feature, computed with
// V_WMMA_F32_16X16X4_F32 (exact fp32: one-hot products are 0 or the feature
// value, f32 accumulate). Each wave owns a private 32x128 accumulator
// (2 M-tiles x 8 N-tiles of v8f = 128 VGPRs) and streams disjoint 4-point
// chunks. Hot loop is branch-free with a scalar (SGPR) trip counter;
// feature loads are non-temporal; prefetch runs 2 iterations ahead.
// Reduction: wave regs -> LDS ds_add_f32 -> global_atomic_add_f32 -> divide.
// ---------------------------------------------------------------------------

typedef float v2f __attribute__((ext_vector_type(2)));
typedef float v8f __attribute__((ext_vector_type(8)));

#define FEAT_D 128   // feature dim (reference: D=128)
#define NSEG   32    // segments    (reference: B=32)
#define SEGSUM_BLOCKS 512
#define SEGSUM_THREADS 256
#define WAVES_PER_BLOCK (SEGSUM_THREADS / 32)

// ---- zero the workspace accumulators (d_ws is poisoned, not zeroed) -------
__global__ void gmp_zero_ws(float* __restrict__ ws, int n) {
  int i = blockIdx.x * blockDim.x + threadIdx.x;
  if (i < n) ws[i] = 0.0f;
}

// ---- main pass: WMMA one-hot segment sum ----------------------------------
__global__ __launch_bounds__(SEGSUM_THREADS)
void gmp_segsum_wmma(const float* __restrict__ feat,   // [npts, 128]
                     const int*   __restrict__ pidx,   // [npts]
                     float* __restrict__ wsum,         // [32*128] zeroed
                     float* __restrict__ wcnt,         // [32]     zeroed
                     long long npts) {
  const int lane = threadIdx.x & 31;     // wave32
  const int col  = lane & 15;            // N-index within a 16-wide tile
  const int half = lane >> 4;            // 0: K rows {0,1}, 1: K rows {2,3}

  // Force the wave id scalar so the loop counter lives in SGPRs and the
  // trip test lowers to s_cmp/s_cbranch instead of the EXEC-mask idiom.
  const int wv = __builtin_amdgcn_readfirstlane(threadIdx.x >> 5);
  const long long wave   = (long long)blockIdx.x * WAVES_PER_BLOCK + wv;
  const long long nwaves = (long long)gridDim.x * WAVES_PER_BLOCK;
  const long long nfull   = npts >> 2;         // chunks with all 4 rows valid
  const long long nchunks = (npts + 3) >> 2;   // incl. possible partial chunk

  // Per-wave accumulators: acc[m][f] = 16x16 f32 C/D tile for
  // segments [m*16, m*16+16) x features [f*16, f*16+16).
  v8f acc[2][8];
#pragma unroll
  for (int m = 0; m < 2; ++m)
#pragma unroll
    for (int f = 0; f < 8; ++f)
      acc[m][f] = (v8f){0.f, 0.f, 0.f, 0.f, 0.f, 0.f, 0.f, 0.f};

  float cnt0 = 0.0f, cnt1 = 0.0f;   // per-lane count partials (seg=col, 16+col)

  // ---- main loop: no bounds checks, no predication, scalar trip count ----
  long long c = wave;
  for (; c < nfull; c += nwaves) {
    // A layout (16x4 f32): lanes 0-15 hold K=0 (v0) / K=1 (v1); lanes 16-31
    // hold K=2 / K=3.  B layout (4x16 f32) mirrors it with N = lane%16.
    const long long r0 = (c << 2) + 2 * half;      // even -> int2 is 8B aligned
    const int2 i01 = *(const int2*)(pidx + r0);    // one global_load_b64
    const float* f0 = feat + (size_t)r0 * FEAT_D;

    // Branchless prefetch two iterations ahead (clamped -> always valid).
    long long rn = ((c + 2 * nwaves) << 2) + 2 * half;
    rn = (rn < npts) ? rn : 0;                      // v_cndmask, no branch
    __builtin_prefetch(feat + (size_t)rn * FEAT_D + col, 0, 0);

    // B tiles: rows r0, r0+1 at column col (read-once stream -> TH_NT).
    v2f b[8];
#pragma unroll
    for (int f = 0; f < 8; ++f) {
      b[f].x = __builtin_nontemporal_load(&f0[f * 16 + col]);
      b[f].y = __builtin_nontemporal_load(&f0[FEAT_D + f * 16 + col]);
    }

#pragma unroll
    for (int m = 0; m < 2; ++m) {
      const int seg = m * 16 + col;
      v2f a;
      a.x = (i01.x == seg) ? 1.0f : 0.0f;
      a.y = (i01.y == seg) ? 1.0f : 0.0f;
      if (m == 0) cnt0 += a.x + a.y; else cnt1 += a.x + a.y;
#pragma unroll
      for (int f = 0; f < 8; ++f) {
        acc[m][f] = __builtin_amdgcn_wmma_f32_16x16x4_f32(
            false, a, false, b[f], (short)0, acc[m][f], false, false);
      }
    }
  }

  // ---- tail: at most one partial chunk per wave (empty when npts%4==0) ----
  for (; c < nchunks; c += nwaves) {
    const long long r0 = (c << 2) + 2 * half;
    const long long r1 = r0 + 1;
    const bool ok0 = (r0 < npts), ok1 = (r1 < npts);
    const int i0 = ok0 ? pidx[r0] : -1;
    const int i1 = ok1 ? pidx[r1] : -1;
    const float* f0 = feat + (size_t)r0 * FEAT_D;

    v2f b[8];
#pragma unroll
    for (int f = 0; f < 8; ++f) {
      b[f].x = ok0 ? f0[f * 16 + col] : 0.0f;
      b[f].y = ok1 ? f0[FEAT_D + f * 16 + col] : 0.0f;
    }
#pragma unroll
    for (int m = 0; m < 2; ++m) {
      const int seg = m * 16 + col;
      v2f a;
      a.x = (i0 == seg) ? 1.0f : 0.0f;
      a.y = (i1 == seg) ? 1.0f : 0.0f;
      if (m == 0) cnt0 += a.x + a.y; else cnt1 += a.x + a.y;
#pragma unroll
      for (int f = 0; f < 8; ++f) {
        acc[m][f] = __builtin_amdgcn_wmma_f32_16x16x4_f32(
            false, a, false, b[f], (short)0, acc[m][f], false, false);
      }
    }
  }

  // ---- block-level reduction in LDS (ds_add_f32), then global atomics ----
  __shared__ float ssum[NSEG * FEAT_D];   // 16 KB
  __shared__ float scnt[NSEG];
  for (int i = threadIdx.x; i < NSEG * FEAT_D; i += blockDim.x) ssum[i] = 0.0f;
  if (threadIdx.x < NSEG) scnt[threadIdx.x] = 0.0f;
  __syncthreads();

  // C/D tile layout: VGPR r holds element (M = r + 8*half, N = col).
#pragma unroll
  for (int m = 0; m < 2; ++m)
#pragma unroll
    for (int f = 0; f < 8; ++f)
#pragma unroll
      for (int r = 0; r < 8; ++r) {
        const int seg = m * 16 + half * 8 + r;
        const int fc  = f * 16 + col;
        atomicAdd(&ssum[seg * FEAT_D + fc], acc[m][f][r]);
      }
  atomicAdd(&scnt[col],      cnt0);   // both wave halves hit same counter: OK
  atomicAdd(&scnt[16 + col], cnt1);
  __syncthreads();

  for (int i = threadIdx.x; i < NSEG * FEAT_D; i += blockDim.x)
    atomicAdd(&wsum[i], ssum[i]);
  if (threadIdx.x < NSEG) atomicAdd(&wcnt[threadIdx.x], scnt[threadIdx.x]);
}

// ---- finalize: mean = sum / count (0/0 -> NaN, matches reference) ---------
__global__ void gmp_finalize(const float* __restrict__ wsum,
                             const float* __restrict__ wcnt,
                             float* __restrict__ out) {
  int i = blockIdx.x * blockDim.x + threadIdx.x;
  if (i < NSEG * FEAT_D) out[i] = wsum[i] / wcnt[i / FEAT_D];
}

extern "C" void kernel_launch(void* const* d_in, const int* in_sizes, int n_in,
                              void* d_out, int out_size, void* d_ws, size_t ws_size,
                              hipStream_t stream) {
  const float* feat = (const float*)d_in[0];  // [N, 128] fp32 (flat N*128)
  const int*   pidx = (const int*)d_in[1];    // [N] int32
  (void)n_in; (void)out_size; (void)ws_size;

  const long long npts = (long long)in_sizes[1];   // N (in_sizes[0] == N*D)

  float* wsum = (float*)d_ws;                 // 32*128 floats
  float* wcnt = wsum + NSEG * FEAT_D;         // 32 floats
  const int nacc = NSEG * FEAT_D + NSEG;

  gmp_zero_ws<<<(nacc + 255) / 256, 256, 0, stream>>>(wsum, nacc);

  gmp_segsum_wmma<<<SEGSUM_BLOCKS, SEGSUM_THREADS, 0, stream>>>(
      feat, pidx, wsum, wcnt, npts);

  gmp_finalize<<<(NSEG * FEAT_D + 255) / 256, 256, 0, stream>>>(
      wsum, wcnt, (float*)d_out);
}